// TripletLossHardMining_24790551233012
// MI455X (gfx1250) — compile-verified
//
#include <hip/hip_runtime.h>

typedef __attribute__((ext_vector_type(2))) float v2f;
typedef __attribute__((ext_vector_type(8))) float v8f;

#define NROW 4096
#define KDIM 1024
#define NLBL 256
#define TMARGIN 0.3f
#define TBIG 1e30f

// ws layout (bytes):
//   0      hpBits uint[4096]   row hardest_pos (float bits, clamped >= 0)
//   16384  hnBits uint[4096]   row hardest_neg (float bits, init = BIG)
//   32768  ss     float[4096]  row same-label dist sum (diag included)
//   49152  st     float[4096]  row total dist sum
//   65536  dii    float[4096]  diag dist = max(0, 1 - ||e_r||^2)
//   81920  hist   uint[256]    label histogram
//   82944  fsc    float[3]     {trip_sum, ps_sum, ns_sum}
//   82960  usc    uint[3]      {n_trip, pc_sum, nc_sum}
#define WS_HP   0
#define WS_HN   16384
#define WS_SS   32768
#define WS_ST   49152
#define WS_DII  65536
#define WS_HIST 81920
#define WS_FS   82944
#define WS_US   82960

// Fragment loader shared by A (16x4, MxK) and B (4x16, KxN) per CDNA5 ISA
// VGPR layouts: lanes 0-15 cover K={k0,k0+1}, lanes 16-31 cover K={k0+2,k0+3},
// with M (A) / N (B) = lane & 15. One contiguous global_load_b64.
__device__ __forceinline__ v2f tl_frag(const float* __restrict__ E, int row, int k) {
  return *reinterpret_cast<const v2f*>(E + (size_t)row * KDIM + k);
}

__global__ void __launch_bounds__(256)
triplet_init_kernel(char* ws) {
  const int tid = blockIdx.x * blockDim.x + threadIdx.x;
  if (tid < NROW) {
    ((unsigned int*)(ws + WS_HP))[tid] = 0u;                    // bits of 0.0f
    ((unsigned int*)(ws + WS_HN))[tid] = __float_as_uint(TBIG);
    ((float*)(ws + WS_SS))[tid] = 0.0f;
    ((float*)(ws + WS_ST))[tid] = 0.0f;
  }
  if (tid < NLBL) ((unsigned int*)(ws + WS_HIST))[tid] = 0u;
  if (tid < 3) {
    ((float*)(ws + WS_FS))[tid] = 0.0f;
    ((unsigned int*)(ws + WS_US))[tid] = 0u;
  }
}

// One wave per row: ||e_r||^2 (for the diagonal term) + label histogram.
__global__ void __launch_bounds__(128)
triplet_norm_hist_kernel(const float* __restrict__ E,
                         const int* __restrict__ labels,
                         char* __restrict__ ws) {
  const int lane = threadIdx.x & 31;
  const int row  = blockIdx.x * 4 + (threadIdx.x >> 5);
  float s = 0.0f;
#pragma unroll
  for (int j = 0; j < KDIM / 32; ++j) {
    const float v = E[(size_t)row * KDIM + lane + j * 32];
    s += v * v;
  }
#pragma unroll
  for (int m = 1; m < 32; m <<= 1) s += __shfl_xor(s, m, 32);
  if (lane == 0) {
    ((float*)(ws + WS_DII))[row] = fmaxf(1.0f - s, 0.0f);
    atomicAdd(&((unsigned int*)(ws + WS_HIST))[labels[row]], 1u);
  }
}

__global__ void __launch_bounds__(128)
triplet_gemm_mine_kernel(const float* __restrict__ E,
                         const int* __restrict__ labels,
                         char* __restrict__ ws) {
  const int lane   = threadIdx.x & 31;
  const int waveId = threadIdx.x >> 5;
  // 64 blocks: blocks 0-31 -> column half 0, blocks 32-63 -> column half 1.
  // All 4 waves of a block share the column half (lockstep B sweep -> L0 hits).
  const int colHalf = blockIdx.x >> 5;
  const int pairId  = (blockIdx.x & 31) * 4 + waveId;   // 0..127, 32 rows each
  const int rowBase = pairId * 32;
  const int colBase = colHalf * (NROW / 2);
  const int nIdx   = lane & 15;   // M index (A) / N index (B,C)
  const int half   = lane >> 4;   // 0: K+0..1, 1: K+2..3
  const int kOff   = half * 2;

  // Row labels: C/D layout -> VGPR i holds M=i (lanes<16) / M=i+8 (lanes>=16).
  int lr[2][8];
#pragma unroll
  for (int s = 0; s < 2; ++s)
#pragma unroll
    for (int i = 0; i < 8; ++i)
      lr[s][i] = labels[rowBase + s * 16 + half * 8 + i];

  // Minimal mining state: hardest_pos, hardest_neg, same-sum, total-sum.
  float hp[2][8], hn[2][8], ssum[2][8], tsum[2][8];
#pragma unroll
  for (int s = 0; s < 2; ++s)
#pragma unroll
    for (int i = 0; i < 8; ++i) {
      hp[s][i] = 0.0f;  hn[s][i] = TBIG;
      ssum[s][i] = 0.0f; tsum[s][i] = 0.0f;
    }

  for (int cb = 0; cb < NROW / 2; cb += 64) {   // 4 column tiles per chunk
    v8f acc[2][4] = {};
    int lc[4];
#pragma unroll
    for (int t = 0; t < 4; ++t) lc[t] = labels[colBase + cb + t * 16 + nIdx];

    for (int k0 = 0; k0 < KDIM; k0 += 4) {
      const v2f a0 = tl_frag(E, rowBase + nIdx,      k0 + kOff);
      const v2f a1 = tl_frag(E, rowBase + 16 + nIdx, k0 + kOff);
      v2f b[4];
#pragma unroll
      for (int t = 0; t < 4; ++t)
        b[t] = tl_frag(E, colBase + cb + t * 16 + nIdx, k0 + kOff);
#pragma unroll
      for (int t = 0; t < 4; ++t) {
        acc[0][t] = __builtin_amdgcn_wmma_f32_16x16x4_f32(
            false, a0, false, b[t], (short)0, acc[0][t], false, false);
        acc[1][t] = __builtin_amdgcn_wmma_f32_16x16x4_f32(
            false, a1, false, b[t], (short)0, acc[1][t], false, false);
      }
    }

    // Fused epilogue: distance + masked row-wise mining, all in registers.
#pragma unroll
    for (int s = 0; s < 2; ++s)
#pragma unroll
      for (int t = 0; t < 4; ++t) {
        const int c = colBase + cb + t * 16 + nIdx;
#pragma unroll
        for (int i = 0; i < 8; ++i) {
          const int r = rowBase + s * 16 + half * 8 + i;
          const float dist = fmaxf(1.0f - acc[s][t][i], 0.0f);
          const bool same = (lr[s][i] == lc[t]);
          tsum[s][i] += dist;
          if (same) {
            ssum[s][i] += dist;                       // diag included; fixed later
            if (r != c) hp[s][i] = fmaxf(hp[s][i], dist);
          } else {
            hn[s][i] = fminf(hn[s][i], dist);
          }
        }
      }
  }

  // Reduce across the 16 lanes sharing each row (xor masks stay within halves).
#pragma unroll
  for (int m = 1; m <= 8; m <<= 1)
#pragma unroll
    for (int s = 0; s < 2; ++s)
#pragma unroll
      for (int i = 0; i < 8; ++i) {
        hp[s][i] = fmaxf(hp[s][i], __shfl_xor(hp[s][i], m, 32));
        hn[s][i] = fminf(hn[s][i], __shfl_xor(hn[s][i], m, 32));
        ssum[s][i] += __shfl_xor(ssum[s][i], m, 32);
        tsum[s][i] += __shfl_xor(tsum[s][i], m, 32);
      }

  if (nIdx == 0) {    // lanes 0 and 16 own 8 rows of each strip
    unsigned int* hpArr = (unsigned int*)(ws + WS_HP);
    unsigned int* hnArr = (unsigned int*)(ws + WS_HN);
    float*        ssArr = (float*)(ws + WS_SS);
    float*        stArr = (float*)(ws + WS_ST);
#pragma unroll
    for (int s = 0; s < 2; ++s)
#pragma unroll
      for (int i = 0; i < 8; ++i) {
        const int r = rowBase + s * 16 + half * 8 + i;
        // Non-negative float bits are order-isomorphic to uint.
        atomicMax(&hpArr[r], __float_as_uint(hp[s][i]));
        atomicMin(&hnArr[r], __float_as_uint(hn[s][i]));
        atomicAdd(&ssArr[r], ssum[s][i]);
        atomicAdd(&stArr[r], tsum[s][i]);
      }
  }
}

__global__ void __launch_bounds__(128)
triplet_rowreduce_kernel(const int* __restrict__ labels,
                         char* __restrict__ ws) {
  const int r = blockIdx.x * blockDim.x + threadIdx.x;

  float tripV = 0.0f, psV = 0.0f, nsV = 0.0f;
  unsigned int validV = 0u, pcV = 0u, ncV = 0u;
  if (r < NROW) {
    const unsigned int h = ((const unsigned int*)(ws + WS_HIST))[labels[r]];
    pcV = h - 1u;                    // same-label count excluding self
    ncV = (unsigned int)NROW - h;    // different-label count
    const float dii = ((const float*)(ws + WS_DII))[r];
    const float ss  = ((const float*)(ws + WS_SS))[r];
    const float st  = ((const float*)(ws + WS_ST))[r];
    psV = ss - dii;                  // positive dist sum (diag removed)
    nsV = st - ss;                   // negative dist sum
    const bool valid = (pcV > 0u) && (ncV > 0u);
    validV = valid ? 1u : 0u;
    const float hpV = __uint_as_float(((const unsigned int*)(ws + WS_HP))[r]);
    const float hnV = __uint_as_float(((const unsigned int*)(ws + WS_HN))[r]);
    tripV = valid ? fmaxf(hpV - hnV + TMARGIN, 0.0f) : 0.0f;
    if (!valid) { psV = 0.0f; nsV = 0.0f; pcV = 0u; ncV = 0u; }
    if (pcV == 0u) psV = 0.0f;       // no positives -> nothing summed
  }
  // Wave-level pre-reduction, then one atomic per wave per quantity.
#pragma unroll
  for (int m = 1; m < 32; m <<= 1) {
    tripV  += __shfl_xor(tripV, m, 32);
    psV    += __shfl_xor(psV, m, 32);
    nsV    += __shfl_xor(nsV, m, 32);
    validV += __shfl_xor(validV, m, 32);
    pcV    += __shfl_xor(pcV, m, 32);
    ncV    += __shfl_xor(ncV, m, 32);
  }
  if ((threadIdx.x & 31) == 0) {
    float* fsc        = (float*)(ws + WS_FS);
    unsigned int* usc = (unsigned int*)(ws + WS_US);
    atomicAdd(&fsc[0], tripV);
    atomicAdd(&fsc[1], psV);
    atomicAdd(&fsc[2], nsV);
    atomicAdd(&usc[0], validV);
    atomicAdd(&usc[1], pcV);
    atomicAdd(&usc[2], ncV);
  }
}

__global__ void triplet_finalize_kernel(const char* __restrict__ ws,
                                        float* __restrict__ out) {
  if (threadIdx.x == 0 && blockIdx.x == 0) {
    const float* fsc        = (const float*)(ws + WS_FS);
    const unsigned int* usc = (const unsigned int*)(ws + WS_US);
    const float tripSum = fsc[0];
    const float psSum   = fsc[1];
    const float nsSum   = fsc[2];
    const unsigned int ntrip = usc[0];
    const unsigned int pcSum = usc[1];
    const unsigned int ncSum = usc[2];
    const float loss = tripSum / (float)(ntrip > 1u ? ntrip : 1u);
    const float pm   = psSum / (float)(pcSum > 1u ? pcSum : 1u);
    const float nm   = nsSum / (float)(ncSum > 1u ? ncSum : 1u);
    out[0] = loss;
    out[1] = pm;
    out[2] = nm;
    out[3] = nm - pm;
    out[4] = (float)ntrip;
  }
}

extern "C" void kernel_launch(void* const* d_in, const int* in_sizes, int n_in,
                              void* d_out, int out_size, void* d_ws, size_t ws_size,
                              hipStream_t stream) {
  (void)in_sizes; (void)n_in; (void)out_size; (void)ws_size;
  const float* E      = (const float*)d_in[0];
  const int*   labels = (const int*)d_in[1];
  char* ws = (char*)d_ws;

  triplet_init_kernel<<<(NROW + 255) / 256, 256, 0, stream>>>(ws);
  triplet_norm_hist_kernel<<<NROW / 4, 128, 0, stream>>>(E, labels, ws);
  triplet_gemm_mine_kernel<<<64, 128, 0, stream>>>(E, labels, ws);
  triplet_rowreduce_kernel<<<NROW / 128, 128, 0, stream>>>(labels, ws);
  triplet_finalize_kernel<<<1, 32, 0, stream>>>(ws, (float*)d_out);
}